// mLSTM_79843442032663
// MI455X (gfx1250) — compile-verified
//
#include <hip/hip_runtime.h>
#include <math.h>

typedef __attribute__((ext_vector_type(2))) float v2f;
typedef __attribute__((ext_vector_type(4))) float v4f;
typedef __attribute__((ext_vector_type(8))) float v8f;

#define MDIM 4096
#define DDIM 2048

// workspace float layout
#define WS_QT   0
#define WS_KT   4096
#define WS_VT   8192
#define WS_Y    12288
#define WS_SCAL 16384   // [0]=it [1]=ft [2]=ot [3]=denom

// ---------------- gate scalars: it = exp(Wi.x+bi), ft = exp(Wf.x+bf), ot = sigmoid(Wo.x+bo)
__global__ __launch_bounds__(256) void gates_kernel(
    const float* __restrict__ x,
    const float* __restrict__ Wi, const float* __restrict__ bi,
    const float* __restrict__ Wf, const float* __restrict__ bf,
    const float* __restrict__ Wo, const float* __restrict__ bo,
    float* __restrict__ scal)
{
    int tid = threadIdx.x;
    float si = 0.f, sf = 0.f, so = 0.f;
    for (int j = tid; j < DDIM; j += 256) {
        float xv = x[j];
        si += Wi[j] * xv; sf += Wf[j] * xv; so += Wo[j] * xv;
    }
    for (int off = 16; off > 0; off >>= 1) {
        si += __shfl_down(si, off); sf += __shfl_down(sf, off); so += __shfl_down(so, off);
    }
    __shared__ float red[8][3];
    int wid = tid >> 5, lane = tid & 31;
    if (lane == 0) { red[wid][0] = si; red[wid][1] = sf; red[wid][2] = so; }
    __syncthreads();
    if (tid == 0) {
        float ti = 0.f, tf = 0.f, to = 0.f;
        for (int w = 0; w < 8; w++) { ti += red[w][0]; tf += red[w][1]; to += red[w][2]; }
        scal[0] = expf(ti + bi[0]);
        scal[1] = expf(tf + bf[0]);
        scal[2] = 1.f / (1.f + expf(-(to + bo[0])));
    }
}

// ---------------- q/k/v GEMVs via V_WMMA_F32_16X16X4_F32; also n = ft*n_prev + it*k
// One wave (32 lanes) per 16-row tile. A = 16x4 fragment of W, B column 0 = x chunk.
// D column 0 lives in lane 0 (rows 0..7 in acc[0..7]) and lane 16 (rows 8..15).
__global__ __launch_bounds__(32) void qkv_wmma_kernel(
    const float* __restrict__ x,
    const float* __restrict__ Wq, const float* __restrict__ bq,
    const float* __restrict__ Wk, const float* __restrict__ bk,
    const float* __restrict__ Wv, const float* __restrict__ bV,
    const float* __restrict__ n_prev, const float* __restrict__ scal,
    float* __restrict__ qt, float* __restrict__ kt, float* __restrict__ vt,
    float* __restrict__ n_out)
{
    const int lane = threadIdx.x;
    const int rowBase = blockIdx.x * 16;
    const int row = rowBase + (lane & 15);
    const int koff = (lane >> 4) << 1;            // 0 for lanes 0-15, 2 for lanes 16-31
    const float* __restrict__ wq = Wq + (size_t)row * DDIM;
    const float* __restrict__ wk = Wk + (size_t)row * DDIM;
    const float* __restrict__ wv = Wv + (size_t)row * DDIM;
    const bool bl = ((lane & 15) == 0);           // only N=0 column of B is live

    v8f aq0 = {}, aq1 = {}, ak0 = {}, ak1 = {}, av0 = {}, av1 = {};

    for (int kc = 0; kc < DDIM; kc += 8) {
        const int k0 = kc + koff;
        const int k1 = kc + 4 + koff;
        float x0 = 0.f, x1 = 0.f, x2 = 0.f, x3 = 0.f;
        if (bl) { x0 = x[k0]; x1 = x[k0 + 1]; x2 = x[k1]; x3 = x[k1 + 1]; }
        v2f b0; b0.x = x0; b0.y = x1;
        v2f b1; b1.x = x2; b1.y = x3;

        v2f a;
        a = *(const v2f*)(wq + k0);
        aq0 = __builtin_amdgcn_wmma_f32_16x16x4_f32(false, a, false, b0, (short)0, aq0, false, false);
        a = *(const v2f*)(wq + k1);
        aq1 = __builtin_amdgcn_wmma_f32_16x16x4_f32(false, a, false, b1, (short)0, aq1, false, false);

        a = *(const v2f*)(wk + k0);
        ak0 = __builtin_amdgcn_wmma_f32_16x16x4_f32(false, a, false, b0, (short)0, ak0, false, false);
        a = *(const v2f*)(wk + k1);
        ak1 = __builtin_amdgcn_wmma_f32_16x16x4_f32(false, a, false, b1, (short)0, ak1, false, false);

        a = *(const v2f*)(wv + k0);
        av0 = __builtin_amdgcn_wmma_f32_16x16x4_f32(false, a, false, b0, (short)0, av0, false, false);
        a = *(const v2f*)(wv + k1);
        av1 = __builtin_amdgcn_wmma_f32_16x16x4_f32(false, a, false, b1, (short)0, av1, false, false);
    }

    v8f q  = aq0 + aq1;
    v8f kk = ak0 + ak1;
    v8f vv = av0 + av1;

    int m0 = (lane == 0) ? 0 : ((lane == 16) ? 8 : -1);
    if (m0 >= 0) {
        const float it = scal[0];
        const float ft = scal[1];
        for (int r = 0; r < 8; r++) {
            const int ro = rowBase + m0 + r;
            const float qv = q[r] + bq[ro];
            const float kv = 0.015625f * (kk[r] + bk[ro]);   // 1/sqrt(4096)
            const float vvv = vv[r] + bV[ro];
            qt[ro] = qv;
            kt[ro] = kv;
            vt[ro] = vvv;
            n_out[ro] = ft * n_prev[ro] + it * kv;
        }
    }
}

// ---------------- denom = max(|n . qt|, 1)
__global__ __launch_bounds__(256) void denom_kernel(
    const float* __restrict__ n, const float* __restrict__ qt, float* __restrict__ scal)
{
    int tid = threadIdx.x;
    float s = 0.f;
    for (int i = tid; i < MDIM; i += 256) s += n[i] * qt[i];
    for (int off = 16; off > 0; off >>= 1) s += __shfl_down(s, off);
    __shared__ float red[8];
    int wid = tid >> 5, lane = tid & 31;
    if (lane == 0) red[wid] = s;
    __syncthreads();
    if (tid == 0) {
        float t = 0.f;
        for (int w = 0; w < 8; w++) t += red[w];
        scal[3] = fmaxf(fabsf(t), 1.0f);
    }
}

// ---------------- fused: Cnew = ft*cp + (it*v_i)*k_j streamed NT; y_i = sum_j Cnew*qt_j
// Block = 256 threads handles 8 full rows. k and qt cached in LDS.
__global__ __launch_bounds__(256) void c_update_kernel(
    const float* __restrict__ cp, const float* __restrict__ qt,
    const float* __restrict__ kt, const float* __restrict__ vt,
    const float* __restrict__ scal, float* __restrict__ Cout, float* __restrict__ y)
{
    __shared__ float sh_k[MDIM];
    __shared__ float sh_q[MDIM];
    __shared__ float wp[8][8];
    const int tid = threadIdx.x;
    for (int j = tid; j < MDIM; j += 256) { sh_k[j] = kt[j]; sh_q[j] = qt[j]; }
    __syncthreads();

    const float it = scal[0];
    const float ft = scal[1];
    const int rowBase = blockIdx.x * 8;
    float acc[8];

    for (int r = 0; r < 8; r++) {
        const int row = rowBase + r;
        const float alpha = it * vt[row];
        const v4f* __restrict__ src = (const v4f*)(cp + (size_t)row * MDIM);
        v4f* __restrict__ dst = (v4f*)(Cout + (size_t)row * MDIM);
        float a = 0.f;
        for (int c = tid; c < MDIM / 4; c += 256) {
            v4f cv = __builtin_nontemporal_load(&src[c]);
            const int j = c * 4;
            v4f nv;
            nv.x = ft * cv.x + alpha * sh_k[j + 0];
            nv.y = ft * cv.y + alpha * sh_k[j + 1];
            nv.z = ft * cv.z + alpha * sh_k[j + 2];
            nv.w = ft * cv.w + alpha * sh_k[j + 3];
            __builtin_nontemporal_store(nv, &dst[c]);
            a += nv.x * sh_q[j + 0] + nv.y * sh_q[j + 1] + nv.z * sh_q[j + 2] + nv.w * sh_q[j + 3];
        }
        acc[r] = a;
    }

    for (int r = 0; r < 8; r++)
        for (int off = 16; off > 0; off >>= 1)
            acc[r] += __shfl_down(acc[r], off);

    const int wid = tid >> 5, lane = tid & 31;
    if (lane == 0)
        for (int r = 0; r < 8; r++) wp[wid][r] = acc[r];
    __syncthreads();
    if (tid < 8) {
        float s = 0.f;
        for (int w = 0; w < 8; w++) s += wp[w][tid];
        y[rowBase + tid] = s;
    }
}

// ---------------- ht = ot * y / denom
__global__ __launch_bounds__(256) void ht_kernel(
    const float* __restrict__ y, const float* __restrict__ scal, float* __restrict__ ht)
{
    const int i = blockIdx.x * 256 + threadIdx.x;
    if (i < MDIM) ht[i] = scal[2] * y[i] / scal[3];
}

extern "C" void kernel_launch(void* const* d_in, const int* in_sizes, int n_in,
                              void* d_out, int out_size, void* d_ws, size_t ws_size,
                              hipStream_t stream)
{
    (void)in_sizes; (void)n_in; (void)out_size; (void)ws_size;
    const float* x      = (const float*)d_in[0];
    const float* cp     = (const float*)d_in[1];
    const float* n_prev = (const float*)d_in[2];
    const float* Wq = (const float*)d_in[3];  const float* bq = (const float*)d_in[4];
    const float* Wk = (const float*)d_in[5];  const float* bk = (const float*)d_in[6];
    const float* Wv = (const float*)d_in[7];  const float* bV = (const float*)d_in[8];
    const float* Wi = (const float*)d_in[9];  const float* bi = (const float*)d_in[10];
    const float* Wf = (const float*)d_in[11]; const float* bf = (const float*)d_in[12];
    const float* Wo = (const float*)d_in[13]; const float* bo = (const float*)d_in[14];

    float* out   = (float*)d_out;
    float* ht    = out;                              // 4096
    float* Cout  = out + MDIM;                       // 4096*4096
    float* n_out = out + MDIM + (size_t)MDIM * MDIM; // 4096

    float* ws   = (float*)d_ws;
    float* qt   = ws + WS_QT;
    float* kt   = ws + WS_KT;
    float* vt   = ws + WS_VT;
    float* y    = ws + WS_Y;
    float* scal = ws + WS_SCAL;

    gates_kernel<<<1, 256, 0, stream>>>(x, Wi, bi, Wf, bf, Wo, bo, scal);
    qkv_wmma_kernel<<<MDIM / 16, 32, 0, stream>>>(x, Wq, bq, Wk, bk, Wv, bV,
                                                  n_prev, scal, qt, kt, vt, n_out);
    denom_kernel<<<1, 256, 0, stream>>>(n_out, qt, scal);
    c_update_kernel<<<MDIM / 8, 256, 0, stream>>>(cp, qt, kt, vt, scal, Cout, y);
    ht_kernel<<<MDIM / 256, 256, 0, stream>>>(y, scal, ht);
}